// SparseDepthMapper_39281770889515
// MI455X (gfx1250) — compile-verified
//
#include <hip/hip_runtime.h>
#include <stdint.h>

// Grid geometry from the reference: CELL=0.1, MAP=40 -> N_CELLS=401, GRID=400, SHIFT=200
#define NGRID   400
#define SHIFT_F 200.0f
#define CELL_F  0.1f

#define BLOCK           256
#define PTS_PER_TILE    1024                    // points per tile
#define FLOATS_PER_TILE (PTS_PER_TILE * 3)      // 3072 floats = 12 KB
#define CHUNK_FLOATS    1024                    // 256 threads * 4 floats per async b128

__global__ void zero_out_kernel(float4* __restrict__ out4, int n4) {
    int i = blockIdx.x * blockDim.x + threadIdx.x;
    if (i < n4) out4[i] = make_float4(0.f, 0.f, 0.f, 0.f);
}

// Low 32 bits of the flat address of an LDS object are the DS byte offset
// (ISA 10.2: LDS aperture).
__device__ __forceinline__ uint32_t lds_off_of(const void* p) {
    return (uint32_t)(uintptr_t)p;
}

// CDNA5 async copy: copies 16 bytes from global address gsrc to LDS offset lds_off, tracked by ASYNCcnt.
__device__ __forceinline__ void async_copy_b128(uint32_t lds_off, const float* gsrc) {
    asm volatile("global_load_async_to_lds_b128 %0, %1, off"
                 :: "v"(lds_off), "v"(gsrc)
                 : "memory");
}

__device__ __forceinline__ void wait_asynccnt_0() {
    asm volatile("s_wait_asynccnt 0" ::: "memory");
}
// <=3 outstanding: with FIFO completion this means the previous tile's 3 ops are done.
__device__ __forceinline__ void wait_asynccnt_3() {
    asm volatile("s_wait_asynccnt 3" ::: "memory");
}

__device__ __forceinline__ void add_point(float x, float ny, float z,
                                          float* __restrict__ out) {
    bool hm = (ny > 0.0f) && (ny < 1.0f);
    // Match jnp: v / 0.1 + 200, round half-to-even (v_rndne_f32), int cast, bounds test.
    float fz = z / CELL_F + SHIFT_F;
    float fx = x / CELL_F + SHIFT_F;
    int iz = (int)rintf(fz);
    int ix = (int)rintf(fx);
    bool ib = (iz >= 0) && (iz < NGRID) && (ix >= 0) && (ix < NGRID);
    if (hm && ib) {
        atomicAdd(out + iz * NGRID + ix, 1.0f);   // global_atomic_add_f32 -> L2
    }
}

__global__ void __launch_bounds__(BLOCK)
scatter_kernel(const float* __restrict__ in, float* __restrict__ out,
               int nPoints, int fullTiles) {
    __shared__ __align__(16) float buf[2][FLOATS_PER_TILE];

    const int tid = threadIdx.x;

    // ---- tail: remainder points (< PTS_PER_TILE) via direct loads, block 0 only ----
    if (blockIdx.x == 0) {
        for (int p = fullTiles * PTS_PER_TILE + tid; p < nPoints; p += BLOCK) {
            float x  = in[3 * p + 0];
            float ny = -in[3 * p + 1];
            float z  = in[3 * p + 2];
            add_point(x, ny, z, out);
        }
    }

    // ---- main: async-staged, double-buffered tiles ----
    auto issue_tile = [&](int tile, int which) {
        const float* gbase = in + (size_t)tile * FLOATS_PER_TILE;
        uint32_t lbase = lds_off_of(&buf[which][0]);
#pragma unroll
        for (int c = 0; c < 3; ++c) {
            int off = c * CHUNK_FLOATS + tid * 4;          // 16B-aligned everywhere
            async_copy_b128(lbase + (uint32_t)off * 4u, gbase + off);
        }
    };

    auto process_tile = [&](int which) {
#pragma unroll
        for (int k = 0; k < PTS_PER_TILE / BLOCK; ++k) {
            int p = tid + k * BLOCK;                       // stride-3 -> conflict-free b32
            float x  = buf[which][3 * p + 0];
            float ny = -buf[which][3 * p + 1];
            float z  = buf[which][3 * p + 2];
            add_point(x, ny, z, out);
        }
    };

    const int t0     = blockIdx.x;
    const int stride = gridDim.x;
    const int myCount = (fullTiles > t0) ? ((fullTiles - 1 - t0) / stride + 1) : 0;

    if (myCount > 0) issue_tile(t0, 0);

    for (int i = 0; i < myCount; ++i) {
        if (i + 1 < myCount) {
            issue_tile(t0 + (i + 1) * stride, (i + 1) & 1); // prefetch next tile
            wait_asynccnt_3();                              // previous tile finished
        } else {
            wait_asynccnt_0();
        }
        __syncthreads();            // other waves' copies done too (each waited its own)
        process_tile(i & 1);
        __syncthreads();            // all reads done before buffer is re-filled
    }
}

extern "C" void kernel_launch(void* const* d_in, const int* in_sizes, int n_in,
                              void* d_out, int out_size, void* d_ws, size_t ws_size,
                              hipStream_t stream) {
    const float* in = (const float*)d_in[0];
    float* out = (float*)d_out;

    int nFloats   = in_sizes[0];            // 24,000,000
    int nPoints   = nFloats / 3;            // 8,000,000
    int fullTiles = nPoints / PTS_PER_TILE; // 7812

    // 1) zero the histogram (harness poisons d_out)
    int n4 = out_size / 4;                  // 160000 floats -> 40000 float4
    zero_out_kernel<<<(n4 + BLOCK - 1) / BLOCK, BLOCK, 0, stream>>>((float4*)out, n4);

    // 2) streaming scatter-add
    int grid = 2048;                        // oversubscribe WGPs; ~4 tiles/block
    scatter_kernel<<<grid, BLOCK, 0, stream>>>(in, out, nPoints, fullTiles);
}